// Mutil_Scale_Attention_14963666059856
// MI455X (gfx1250) — compile-verified
//
#include <hip/hip_runtime.h>
#include <hip/hip_bf16.h>
#include <stdint.h>

// ---------------- problem constants ----------------
#define BB   2
#define LL   2046
#define DD   1024
#define DKK  64
#define HH   16
#define GHH  8
#define LHH  8
#define WSS  3
#define NWW  682            // LL / WSS
#define SCALE 0.125f        // 1/sqrt(DK)
#define MAXV  1000000000.0f

typedef __attribute__((ext_vector_type(2))) float    v2f;
typedef __attribute__((ext_vector_type(8))) float    v8f;
typedef __attribute__((ext_vector_type(4))) uint32_t u32x4;
typedef __attribute__((ext_vector_type(8))) uint32_t u32x8;

// D = A(16x4,f32) * B(4x16,f32) + C(16x16,f32), full f32 precision
__device__ __forceinline__ v8f wmma4(v2f a, v2f b, v8f c) {
  return __builtin_amdgcn_wmma_f32_16x16x4_f32(
      /*neg_a=*/false, a, /*neg_b=*/false, b,
      /*c_mod=*/(short)0, c, /*reuse_a=*/false, /*reuse_b=*/false);
}

__device__ __forceinline__ float grp16_max(float v) {
#pragma unroll
  for (int m = 8; m >= 1; m >>= 1) v = fmaxf(v, __shfl_xor(v, m, 32));
  return v;
}
__device__ __forceinline__ float grp16_sum(float v) {
#pragma unroll
  for (int m = 8; m >= 1; m >>= 1) v += __shfl_xor(v, m, 32);
  return v;
}

// ---------------------------------------------------------------
// TDM: DMA a contiguous run of f32 (one "row" tile) from global to LDS.
// D# group0: count=1 | lds_addr | global_addr[56:0] | type=2
// D# group1: data_size=4B | tensor_dim0 = n_valid (HW zero-fills past it)
//            tensor_dim1=1 | tile_dim0 = n_tile | tile_dim1=1
// ---------------------------------------------------------------
__device__ __forceinline__ void tdm_load_row(uint32_t lds_off, const void* gptr,
                                             uint32_t n_valid, uint32_t n_tile) {
  const uint64_t ga = (uint64_t)(uintptr_t)gptr;
  u32x4 g0;
  g0[0] = 1u;                                                   // count=1
  g0[1] = lds_off;                                              // lds_addr
  g0[2] = (uint32_t)ga;                                         // global_addr lo
  g0[3] = (uint32_t)((ga >> 32) & 0x01FFFFFFu) | (2u << 30);    // ga hi | type=2
  u32x8 g1;
  g1[0] = 2u << 16;                                             // data_size = 4B
  g1[1] = (n_valid & 0xFFFFu) << 16;                            // tensor_dim0 lo16
  g1[2] = ((n_valid >> 16) & 0xFFFFu) | (1u << 16);             // td0 hi16 | tensor_dim1=1
  g1[3] = (n_tile & 0xFFFFu) << 16;                             // tile_dim0
  g1[4] = 1u;                                                   // tile_dim1=1, tile_dim2=0
  g1[5] = n_tile;                                               // tensor_dim0_stride lo
  g1[6] = 0u;
  g1[7] = 0u;
  asm volatile("tensor_load_to_lds %0, %1" :: "s"(g0), "s"(g1) : "memory");
}

__device__ __forceinline__ uint32_t lds_offset_of(const void* p) {
  return (uint32_t)(uintptr_t)p;   // generic LDS pointer: addr[31:0] = LDS offset
}

// ---------------------------------------------------------------
// Kernel 1: q = x @ Wq^T + bq, stored as [b][h][l][dk]
// grid: (D/64, ceil(L/16), B), block 128 (4 waves, each a 16-col tile)
// ---------------------------------------------------------------
__global__ void qproj_kernel(const float* __restrict__ x,
                             const float* __restrict__ Wq,
                             const float* __restrict__ bq,
                             float* __restrict__ q) {
  const int lane = threadIdx.x & 31;
  const int wave = threadIdx.x >> 5;
  const int n0   = blockIdx.x * 64 + wave * 16;
  const int row0 = blockIdx.y * 16;
  const int b    = blockIdx.z;
  const int col  = lane & 15;
  const int kd   = (lane < 16) ? 0 : 2;

  const int arow = min(row0 + col, LL - 1);
  const float* xrow = x  + (size_t)(b * LL + arow) * DD;
  const float* wrow = Wq + (size_t)(n0 + col) * DD;

  v8f c = {};
#pragma unroll 4
  for (int i = 0; i < DD / 4; ++i) {
    v2f a  = *(const v2f*)(xrow + 4 * i + kd);
    v2f bf = *(const v2f*)(wrow + 4 * i + kd);
    c = wmma4(a, bf, c);
  }
  const float bias = bq[n0 + col];
  const int h  = n0 >> 6;
  const int dk = (n0 & 63) + col;
  const int rbase = (lane < 16) ? 0 : 8;
#pragma unroll
  for (int r = 0; r < 8; ++r) {
    const int row = row0 + rbase + r;
    if (row < LL)
      q[(((size_t)b * HH + h) * LL + row) * DKK + dk] = c[r] + bias;
  }
}

// ---------------------------------------------------------------
// Kernel 2a: kw[b][h][n][dk] = mean over WS of x[b][3n+j][(GH+h)*DK+dk]
// ---------------------------------------------------------------
__global__ void kwmean_kernel(const float* __restrict__ x, float* __restrict__ kw) {
  const int total = BB * LHH * NWW * DKK;
  int idx = blockIdx.x * blockDim.x + threadIdx.x;
  if (idx >= total) return;
  const int dk = idx & (DKK - 1);
  int t = idx / DKK;
  const int n = t % NWW; t /= NWW;
  const int h = t % LHH;
  const int b = t / LHH;
  const int c = (GHH + h) * DKK + dk;
  const float* base = x + (size_t)(b * LL + 3 * n) * DD + c;
  kw[idx] = (base[0] + base[DD] + base[2 * DD]) * (1.0f / 3.0f);
}

// ---------------------------------------------------------------
// Kernel 2b: per-head projection  dst[b,h,m,:] = S[b,h,m,:] @ W[h]^T + bias[h]
// ---------------------------------------------------------------
__global__ void headproj_kernel(const float* __restrict__ src,
                                long srcBS, long srcHS, int srcRS, int M,
                                const float* __restrict__ W,
                                const float* __restrict__ bias,
                                float* __restrict__ dst,
                                long dstBS, long dstHS) {
  const int lane = threadIdx.x & 31;
  const int wave = threadIdx.x >> 5;
  const int n0   = wave * 16;
  const int row0 = blockIdx.y * 16;
  const int b    = blockIdx.z >> 3;
  const int h    = blockIdx.z & 7;
  const int col  = lane & 15;
  const int kd   = (lane < 16) ? 0 : 2;

  const float* srow = src + b * srcBS + h * srcHS
                      + (size_t)min(row0 + col, M - 1) * srcRS;
  const float* wrow = W + ((size_t)h * DKK + n0 + col) * DKK;

  v8f c = {};
#pragma unroll
  for (int i = 0; i < DKK / 4; ++i) {
    v2f a  = *(const v2f*)(srow + 4 * i + kd);
    v2f bf = *(const v2f*)(wrow + 4 * i + kd);
    c = wmma4(a, bf, c);
  }
  const float bv = bias[h * DKK + n0 + col];
  float* drow = dst + b * dstBS + h * dstHS;
  const int rbase = (lane < 16) ? 0 : 8;
#pragma unroll
  for (int r = 0; r < 8; ++r) {
    const int row = row0 + rbase + r;
    if (row < M)
      drow[(size_t)row * DKK + n0 + col] = c[r] + bv;
  }
}

// ---------------------------------------------------------------
// Kernel 3: attention (global: NK=L, local: NK=NW).
// One wave per 16-row query tile; all 4 waves of a block share (b,h) and the
// key-tile sequence, so K/V 16x64 tiles are DMA'd once per block into LDS via
// TDM (wave 0 issues tensor_load_to_lds, waits TENSORcnt, block barriers).
// grid: (1, 32, B*8), block 128
// ---------------------------------------------------------------
template <bool LOCAL>
__global__ void attention_kernel(const float* __restrict__ q,     // [B][H][L][DK]
                                 const float* __restrict__ kp,    // [B][8][NK][DK]
                                 const float* __restrict__ vp,    // [B][8][NK][DK]
                                 const float* __restrict__ mask,  // [B][L][L]
                                 float* __restrict__ pout,        // [B][8][L][NK]
                                 float* __restrict__ vout,        // [B][H][L][DK]
                                 int NK) {
  __shared__ __align__(16) float ldsP[4][16][16];   // per-wave P tile
  __shared__ __align__(16) float ldsK[16 * DKK];    // shared K tile (rows x 64)
  __shared__ __align__(16) float ldsV[16 * DKK];    // shared V tile (rows x 64)

  const int lane  = threadIdx.x & 31;
  const int wave  = threadIdx.x >> 5;
  const int b     = blockIdx.z >> 3;
  const int h     = blockIdx.z & 7;
  const int qt    = blockIdx.y * 4 + wave;
  const int row0  = qt * 16;
  const int col   = lane & 15;
  const int kd    = (lane < 16) ? 0 : 2;
  const int rbase = (lane < 16) ? 0 : 8;
  const int nkt   = (NK + 15) / 16;
  const int vhead = LOCAL ? (h + 8) : h;

  const float* qhead = q  + (((size_t)b * HH + vhead) * LL) * DKK;
  const float* khead = kp + (((size_t)b * 8 + h) * NK) * DKK;
  const float* vhd   = vp + (((size_t)b * 8 + h) * NK) * DKK;
  const float* mrowb = mask + ((size_t)b * LL) * LL;
  const uint32_t ldsK_off = lds_offset_of(&ldsK[0]);
  const uint32_t ldsV_off = lds_offset_of(&ldsV[0]);

  // preload q A-fragments for the whole DK=64 reduction (reused in both passes)
  const int arow = min(row0 + col, LL - 1);
  const float* qrowp = qhead + (size_t)arow * DKK;
  v2f qa[16];
#pragma unroll
  for (int i = 0; i < 16; ++i) qa[i] = *(const v2f*)(qrowp + 4 * i + kd);

  float mrun[8], srun[8];
#pragma unroll
  for (int r = 0; r < 8; ++r) { mrun[r] = -3.0e38f; srun[r] = 0.0f; }

  // ---------------- pass A: online softmax stats ----------------
  for (int kt = 0; kt < nkt; ++kt) {
    const int key0 = kt * 16;
    const int keyc = min(key0 + col, NK - 1);
    const bool kok = (key0 + col) < NK;

    __syncthreads();  // everyone done reading ldsK from previous tile
    if (wave == 0) {
      tdm_load_row(ldsK_off, khead + (size_t)key0 * DKK,
                   (uint32_t)(NK - key0) * DKK, 16 * DKK);
      __builtin_amdgcn_s_wait_tensorcnt(0);
      __builtin_prefetch(khead + (size_t)min(key0 + 16, NK - 1) * DKK, 0, 2);
    }
    __syncthreads();  // staged K tile visible to all waves

    v8f c = {};
#pragma unroll
    for (int i = 0; i < 16; ++i) {
      v2f bf = *(const v2f*)(&ldsK[col * DKK + 4 * i + kd]);
      c = wmma4(qa[i], bf, c);
    }
#pragma unroll
    for (int r = 0; r < 8; ++r) {
      const int qr = min(row0 + rbase + r, LL - 1);
      float v;
      if (LOCAL) {
        const float* mp = mrowb + (size_t)qr * LL + 3 * keyc;
        const float cnt = mp[0] + mp[1] + mp[2];
        v = (kok && cnt >= 2.0f) ? c[r] * SCALE : -MAXV;
      } else {
        const float mv = mrowb[(size_t)qr * LL + keyc];
        v = (kok && mv != 0.0f) ? c[r] * SCALE : -MAXV;
      }
      const float mt   = grp16_max(v);
      const float mnew = fmaxf(mrun[r], mt);
      const float psum = grp16_sum(__expf(v - mnew));
      srun[r] = srun[r] * __expf(mrun[r] - mnew) + psum;
      mrun[r] = mnew;
    }
  }
  float rinv[8];
#pragma unroll
  for (int r = 0; r < 8; ++r) rinv[r] = 1.0f / srun[r];

  // ---------------- pass B: write P, accumulate P@V ----------------
  v8f acc[4] = {{}, {}, {}, {}};
  float* prow = pout + (((size_t)b * 8 + h) * LL) * NK;

  for (int kt = 0; kt < nkt; ++kt) {
    const int key0 = kt * 16;
    const int keyc = min(key0 + col, NK - 1);
    const bool kok = (key0 + col) < NK;

    __syncthreads();  // prior tile's ldsK/ldsV/ldsP reads complete
    if (wave == 0) {
      tdm_load_row(ldsK_off, khead + (size_t)key0 * DKK,
                   (uint32_t)(NK - key0) * DKK, 16 * DKK);
      tdm_load_row(ldsV_off, vhd + (size_t)key0 * DKK,
                   (uint32_t)(NK - key0) * DKK, 16 * DKK);
      __builtin_amdgcn_s_wait_tensorcnt(0);
    }
    __syncthreads();  // staged K/V tiles visible

    v8f c = {};
#pragma unroll
    for (int i = 0; i < 16; ++i) {
      v2f bf = *(const v2f*)(&ldsK[col * DKK + 4 * i + kd]);
      c = wmma4(qa[i], bf, c);
    }
#pragma unroll
    for (int r = 0; r < 8; ++r) {
      const int qrow = row0 + rbase + r;
      const int qr   = min(qrow, LL - 1);
      float v;
      if (LOCAL) {
        const float* mp = mrowb + (size_t)qr * LL + 3 * keyc;
        const float cnt = mp[0] + mp[1] + mp[2];
        v = (kok && cnt >= 2.0f) ? c[r] * SCALE : -MAXV;
      } else {
        const float mv = mrowb[(size_t)qr * LL + keyc];
        v = (kok && mv != 0.0f) ? c[r] * SCALE : -MAXV;
      }
      const float p = __expf(v - mrun[r]) * rinv[r];
      ldsP[wave][rbase + r][col] = p;
      if (qrow < LL && kok) prow[(size_t)qrow * NK + key0 + col] = p;
    }
    __syncthreads();  // P tile visible (per-wave) before A-fragment reads
#pragma unroll
    for (int j = 0; j < 4; ++j) {
      const int n0 = j * 16;
#pragma unroll
      for (int i = 0; i < 4; ++i) {
        v2f a = *(const v2f*)(&ldsP[wave][col][4 * i + kd]);
        v2f bf;
        bf.x = ldsV[(4 * i + kd)     * DKK + n0 + col];
        bf.y = ldsV[(4 * i + kd + 1) * DKK + n0 + col];
        acc[j] = wmma4(a, bf, acc[j]);
      }
    }
  }

  float* vrow = vout + (((size_t)b * HH + vhead) * LL) * DKK;
#pragma unroll
  for (int j = 0; j < 4; ++j)
#pragma unroll
    for (int r = 0; r < 8; ++r) {
      const int qrow = row0 + rbase + r;
      if (qrow < LL)
        vrow[(size_t)qrow * DKK + j * 16 + col] = acc[j][r];
    }
}

// ---------------------------------------------------------------
extern "C" void kernel_launch(void* const* d_in, const int* in_sizes, int n_in,
                              void* d_out, int out_size, void* d_ws, size_t ws_size,
                              hipStream_t stream) {
  (void)in_sizes; (void)n_in; (void)out_size; (void)ws_size;
  const float* x    = (const float*)d_in[0];
  const float* mask = (const float*)d_in[1];
  const float* Wq   = (const float*)d_in[2];
  const float* bq   = (const float*)d_in[3];
  const float* Wk   = (const float*)d_in[4];
  const float* bk   = (const float*)d_in[5];
  const float* Wv   = (const float*)d_in[6];
  const float* bv   = (const float*)d_in[7];
  float* out = (float*)d_out;
  float* ws  = (float*)d_ws;

  // workspace layout (floats)
  float* q   = ws;                                  // B*H*L*DK
  float* kgp = q   + (size_t)BB * HH  * LL  * DKK;  // B*8*L*DK
  float* vgp = kgp + (size_t)BB * GHH * LL  * DKK;
  float* kw  = vgp + (size_t)BB * GHH * LL  * DKK;  // B*8*NW*DK
  float* klp = kw  + (size_t)BB * LHH * NWW * DKK;
  float* vlp = klp + (size_t)BB * LHH * NWW * DKK;

  // output layout: value_all | pg | pl
  float* vall = out;
  float* pg   = out + (size_t)BB * HH * LL * DKK;
  float* pl   = pg  + (size_t)BB * GHH * LL * (size_t)LL;

  const int LT = (LL + 15) / 16;   // 128
  const int NT = (NWW + 15) / 16;  // 43

  qproj_kernel<<<dim3(DD / 64, LT, BB), 128, 0, stream>>>(x, Wq, bq, q);

  {
    const int total = BB * LHH * NWW * DKK;
    kwmean_kernel<<<(total + 255) / 256, 256, 0, stream>>>(x, kw);
  }

  // global-head K/V projections: S[b,h,m,i] = x[b, m, h*DK+i]
  headproj_kernel<<<dim3(1, LT, BB * 8), 128, 0, stream>>>(
      x, (long)LL * DD, (long)DKK, DD, LL, Wk, bk,
      kgp, (long)GHH * LL * DKK, (long)LL * DKK);
  headproj_kernel<<<dim3(1, LT, BB * 8), 128, 0, stream>>>(
      x, (long)LL * DD, (long)DKK, DD, LL, Wv, bv,
      vgp, (long)GHH * LL * DKK, (long)LL * DKK);
  // local-head projections of window-averaged keys
  headproj_kernel<<<dim3(1, NT, BB * 8), 128, 0, stream>>>(
      kw, (long)LHH * NWW * DKK, (long)NWW * DKK, DKK, NWW,
      Wk + GHH * DKK * DKK, bk + GHH * DKK,
      klp, (long)LHH * NWW * DKK, (long)NWW * DKK);
  headproj_kernel<<<dim3(1, NT, BB * 8), 128, 0, stream>>>(
      kw, (long)LHH * NWW * DKK, (long)NWW * DKK, DKK, NWW,
      Wv + GHH * DKK * DKK, bv + GHH * DKK,
      vlp, (long)LHH * NWW * DKK, (long)NWW * DKK);

  attention_kernel<false><<<dim3(1, LT / 4, BB * 8), 128, 0, stream>>>(
      q, kgp, vgp, mask, pg, vall, LL);
  attention_kernel<true><<<dim3(1, LT / 4, BB * 8), 128, 0, stream>>>(
      q, klp, vlp, mask, pl, vall, NWW);
}